// DeepSTNModel_75058848465280
// MI455X (gfx1250) — compile-verified
//
#include <hip/hip_runtime.h>
#include <math.h>

// ---------------------------------------------------------------------------
// DeepSTN forward for MI455X (gfx1250), fp32 end-to-end.
// Dominant cost: 4x streaming of the 537MB plus_w weight matrix from HBM
// (~92us at 23.3 TB/s). GEMMs use V_WMMA_F32_16X16X4_F32 so compute stays
// well above the ~373 TFLOP/s needed to remain HBM-bound. Activations for
// the plus GEMM are stored K-pair-interleaved so every B fragment is one
// b64 load (3 vmem per 2 WMMA in the hot loop).
// ---------------------------------------------------------------------------

namespace {
constexpr int NB  = 32;            // batch
constexpr int HH  = 32, WW = 16, HW = 512;
constexpr int CC  = 64;            // cpt channels
constexpr int PH  = 34, PW = 18, PCH = PH * PW;   // padded plane, channel stride 612
constexpr int KPLUS = 32768, NPLUS = 4096, KSPLIT = 8;
constexpr float FEPS = 1e-5f;
}

typedef __attribute__((ext_vector_type(2))) float v2f;
typedef __attribute__((ext_vector_type(8))) float v8f;

// D = A(16x4) * B(4x16) + C, fp32 WMMA.
// Lane layout (ISA 7.12.2): A: lanes0-15 M=lane hold K=k0,k0+1; lanes16-31
// hold K=k0+2,k0+3. B: mirrored on N. C/D: VGPR r = rows M=r / M=r+8.
__device__ __forceinline__ v8f wmma4(v2f a, v2f b, v8f c) {
  return __builtin_amdgcn_wmma_f32_16x16x4_f32(false, a, false, b,
                                               (short)0, c, false, false);
}

// ------------------------------ fill ---------------------------------------
__global__ void k_fill_zero(float* __restrict__ p, int n) {
  int i = blockIdx.x * blockDim.x + threadIdx.x;
  if (i < n) p[i] = 0.f;
}

// ------------------------------ head ---------------------------------------
// Per pixel: c/p/t 1x1 convs, tx path, poi_time, relu->bn1, then 201->64 1x1.
__global__ __launch_bounds__(256) void k_head(
    const float* __restrict__ x,
    const float* __restrict__ wc, const float* __restrict__ bc,
    const float* __restrict__ wp, const float* __restrict__ bp,
    const float* __restrict__ wt, const float* __restrict__ bt,
    const float* __restrict__ tmw, const float* __restrict__ tmb,
    const float* __restrict__ tfw, const float* __restrict__ tfb,
    const float* __restrict__ pw,  const float* __restrict__ pb,
    const float* __restrict__ g1,  const float* __restrict__ b1,
    const float* __restrict__ m1,  const float* __restrict__ v1,
    const float* __restrict__ w1,  const float* __restrict__ bb1,
    float* __restrict__ cpt) {
  int pix = blockIdx.x * blockDim.x + threadIdx.x;   // 16384 threads
  int b = pix >> 9, hw = pix & 511;
  const float* xb = x + (size_t)b * 65 * HW + hw;

  float xin[22];
#pragma unroll
  for (int i = 0; i < 22; ++i) xin[i] = xb[(size_t)i * HW];
  float poi[12];
#pragma unroll
  for (int i = 0; i < 12; ++i) poi[i] = xb[(size_t)(22 + i) * HW];
  float txa;
  {
    float tin[31];
#pragma unroll
    for (int i = 0; i < 31; ++i) tin[i] = xb[(size_t)(34 + i) * HW];
    txa = tfb[0];
    for (int j = 0; j < 28; ++j) {
      float s = tmb[j];
#pragma unroll
      for (int k = 0; k < 31; ++k) s += tmw[j * 31 + k] * tin[k];
      txa += tfw[j] * fmaxf(s, 0.f);
    }
  }
  float txv = fmaxf(txa, 0.f);

  float acc[64];
#pragma unroll
  for (int o = 0; o < 64; ++o) acc[o] = bb1[o];

  // j = concat-channel index (0..200): relu then bn1, feed conv1.
  for (int j = 0; j < 64; ++j) {            // c block
    float v = bc[j];
#pragma unroll
    for (int k = 0; k < 6; ++k) v += wc[j * 6 + k] * xin[k];
    float sc = g1[j] * rsqrtf(v1[j] + FEPS);
    v = fmaxf(v, 0.f) * sc + (b1[j] - m1[j] * sc);
#pragma unroll
    for (int o = 0; o < 64; ++o) acc[o] += w1[o * 201 + j] * v;
  }
  for (int jj = 0; jj < 64; ++jj) {         // p block
    int j = 64 + jj;
    float v = bp[jj];
#pragma unroll
    for (int k = 0; k < 8; ++k) v += wp[jj * 8 + k] * xin[6 + k];
    float sc = g1[j] * rsqrtf(v1[j] + FEPS);
    v = fmaxf(v, 0.f) * sc + (b1[j] - m1[j] * sc);
#pragma unroll
    for (int o = 0; o < 64; ++o) acc[o] += w1[o * 201 + j] * v;
  }
  for (int jj = 0; jj < 64; ++jj) {         // t block
    int j = 128 + jj;
    float v = bt[jj];
#pragma unroll
    for (int k = 0; k < 8; ++k) v += wt[jj * 8 + k] * xin[14 + k];
    float sc = g1[j] * rsqrtf(v1[j] + FEPS);
    v = fmaxf(v, 0.f) * sc + (b1[j] - m1[j] * sc);
#pragma unroll
    for (int o = 0; o < 64; ++o) acc[o] += w1[o * 201 + j] * v;
  }
  for (int jj = 0; jj < 9; ++jj) {          // poi_time block
    int j = 192 + jj;
    float v = pb[jj];
#pragma unroll
    for (int k = 0; k < 12; ++k) v += pw[jj * 12 + k] * (txv * poi[k]);
    float sc = g1[j] * rsqrtf(v1[j] + FEPS);
    v = fmaxf(v, 0.f) * sc + (b1[j] - m1[j] * sc);
#pragma unroll
    for (int o = 0; o < 64; ++o) acc[o] += w1[o * 201 + j] * v;
  }
#pragma unroll
  for (int o = 0; o < 64; ++o)
    cpt[((size_t)b * CC + o) * HW + hw] = acc[o];
}

// ------------------------------ prep ---------------------------------------
// apad interior = relu(bn1rp(cpt));
// z2tp = bn1rp(relu(cpt)) transposed & K-pair interleaved:
//   z2tp[((k>>1)*NB + b)*2 + (k&1)] = act[b][k],  k = c*512 + hw.
__global__ __launch_bounds__(256) void k_prep(
    const float* __restrict__ cpt,
    const float* __restrict__ g, const float* __restrict__ bb,
    const float* __restrict__ m, const float* __restrict__ vv,
    float* __restrict__ apad, float* __restrict__ z2tp) {
  int i = blockIdx.x * blockDim.x + threadIdx.x;   // 1,048,576 threads
  int b = i >> 15, c = (i >> 9) & 63, hw = i & 511;
  int h = hw >> 4, w = hw & 15;
  float v = cpt[i];
  float sc = g[c] * rsqrtf(vv[c] + FEPS);
  float tt = bb[c] - m[c] * sc;
  apad[(size_t)(b * CC + c) * PCH + (size_t)(h + 1) * PW + (w + 1)] =
      fmaxf(v * sc + tt, 0.f);
  int k = c * HW + hw;
  z2tp[(size_t)((k >> 1) * NB + b) * 2 + (k & 1)] = fmaxf(v, 0.f) * sc + tt;
}

// --------------------------- plus GEMM -------------------------------------
// C(4096 x 32) = plus_w(4096 x 32768) * act(32768 x 32), K split 8 ways.
// A rows = weight rows -> per-lane contiguous HBM streaming (b64 loads).
// B fragments: one b64 load each thanks to the K-pair interleaved layout.
__global__ __launch_bounds__(32) void k_plus_gemm(
    const float* __restrict__ pwf,   // plus_w reshaped (4096, 32768)
    const float* __restrict__ z2tp,  // (16384, 32, 2) K-pair interleaved
    float* __restrict__ part) {      // (KSPLIT, 4096, 32)
  int lane = threadIdx.x & 31, hi = lane >> 4, lo = lane & 15;
  int f0 = blockIdx.x * 16;          // feature tile (256)
  int ks = blockIdx.y;               // K split    (8)
  const float* Wr = pwf + (size_t)(f0 + lo) * KPLUS;
  int kbeg = ks * (KPLUS / KSPLIT), kend = kbeg + (KPLUS / KSPLIT);

  v8f acc0 = {0, 0, 0, 0, 0, 0, 0, 0};
  v8f acc1 = {0, 0, 0, 0, 0, 0, 0, 0};
  for (int kb = kbeg; kb < kend; kb += 32) {
    __builtin_prefetch(Wr + kb + 256, 0, 0);   // global_prefetch_b8 ahead
#pragma unroll
    for (int u = 0; u < 8; ++u) {
      int k = kb + u * 4 + 2 * hi;             // even; pair (k,k+1) adjacent
      v2f a = *(const v2f*)(Wr + k);           // weight stream, b64 per lane
      const float* bp_ = z2tp + (size_t)(k >> 1) * (NB * 2);
      v2f bA = *(const v2f*)(bp_ + lo * 2);         // batches 0..15
      v2f bB = *(const v2f*)(bp_ + 32 + lo * 2);    // batches 16..31
      acc0 = wmma4(a, bA, acc0);
      acc1 = wmma4(a, bB, acc1);
    }
  }
#pragma unroll
  for (int r = 0; r < 8; ++r) {
    int f = f0 + r + 8 * hi;
    part[((size_t)ks * NPLUS + f) * NB + lo]      = acc0[r];
    part[((size_t)ks * NPLUS + f) * NB + 16 + lo] = acc1[r];
  }
}

__global__ __launch_bounds__(256) void k_plus_reduce(
    const float* __restrict__ part, const float* __restrict__ pbias,
    float* __restrict__ z2out) {     // (4096, 32)
  int i = blockIdx.x * blockDim.x + threadIdx.x;   // 131072
  int f = i >> 5;
  float s = pbias[f];
#pragma unroll
  for (int ks = 0; ks < KSPLIT; ++ks) s += part[(size_t)ks * NPLUS * NB + i];
  z2out[i] = s;
}

// --------------------------- conv3x3 (WMMA im2col) -------------------------
// One wave per (batch, image-row, 16-out-channel tile); K = (r,s,c), 144 WMMAs.
__global__ __launch_bounds__(32) void k_conv3x3(
    const float* __restrict__ inpad,  // (NB, 64, 34, 18) zero-padded
    const float* __restrict__ wgt,    // (OC, 64, 3, 3)
    const float* __restrict__ bias,   // (OC)
    float* __restrict__ outp,         // (NB, 64, 32, 16) stride-64 layout
    const float* __restrict__ resid,  // nullptr or same layout as outp
    int OC) {
  int lane = threadIdx.x & 31, hi = lane >> 4, lo = lane & 15;
  int bh = blockIdx.x;                // 1024 = NB*HH
  int b = bh >> 5, h = bh & 31;
  int o0 = blockIdx.y * 16;
  int oA = o0 + lo;                   // A-matrix row (out channel)

  v8f acc;
#pragma unroll
  for (int r = 0; r < 8; ++r) {
    int oo = o0 + r + 8 * hi;
    acc[r] = (oo < OC) ? bias[oo] : 0.f;
  }
  for (int rs = 0; rs < 9; ++rs) {
    int rr = rs / 3, ss = rs % 3;
    const float* prow = inpad + (size_t)b * CC * PCH + (size_t)(h + rr) * PW + ss;
    const float* wrow = (oA < OC) ? (wgt + (size_t)oA * CC * 9 + rs) : nullptr;
#pragma unroll 8
    for (int c0 = 0; c0 < CC; c0 += 4) {
      int c = c0 + 2 * hi;
      v2f a, bf;
      a.x = wrow ? wrow[(size_t)c * 9] : 0.f;
      a.y = wrow ? wrow[(size_t)(c + 1) * 9] : 0.f;
      bf.x = prow[(size_t)c * PCH + lo];        // 16 contiguous pixels/lane row
      bf.y = prow[(size_t)(c + 1) * PCH + lo];
      acc = wmma4(a, bf, acc);
    }
  }
#pragma unroll
  for (int r = 0; r < 8; ++r) {
    int oo = o0 + r + 8 * hi;
    if (oo < OC) {
      size_t idx = (((size_t)b * CC + oo) * HH + h) * WW + lo;
      float v = acc[r];
      if (resid) v += resid[idx];
      outp[idx] = v;
    }
  }
}

// --------------------------- z3 build --------------------------------------
// z3pad interior = relu(bn2rp(concat(z1[0:56], z2[0:8]))).
__global__ __launch_bounds__(256) void k_z3(
    const float* __restrict__ z1, const float* __restrict__ z2out,
    const float* __restrict__ g, const float* __restrict__ bb,
    const float* __restrict__ m, const float* __restrict__ vv,
    float* __restrict__ z3pad) {
  int i = blockIdx.x * blockDim.x + threadIdx.x;   // 1,048,576
  int b = i >> 15, o = (i >> 9) & 63, hw = i & 511;
  int h = hw >> 4, w = hw & 15;
  float val = (o < 56) ? z1[i]
                       : z2out[(size_t)((o - 56) * HW + hw) * NB + b];
  float sc = g[o] * rsqrtf(vv[o] + FEPS);
  val = fmaxf(val * sc + (bb[o] - m[o] * sc), 0.f);
  z3pad[(size_t)(b * CC + o) * PCH + (size_t)(h + 1) * PW + (w + 1)] = val;
}

// ------------------------------ tail ---------------------------------------
__global__ __launch_bounds__(256) void k_tail(
    const float* __restrict__ cpt,
    const float* __restrict__ g, const float* __restrict__ bb,
    const float* __restrict__ m, const float* __restrict__ vv,
    const float* __restrict__ w2, const float* __restrict__ b2,
    float* __restrict__ out) {
  int pix = blockIdx.x * blockDim.x + threadIdx.x;  // 16384
  int b = pix >> 9, hw = pix & 511;
  float a0 = b2[0], a1 = b2[1];
  for (int o = 0; o < 64; ++o) {
    float v = cpt[((size_t)b * CC + o) * HW + hw];
    float sc = g[o] * rsqrtf(vv[o] + FEPS);
    float u = fmaxf(v, 0.f) * sc + (bb[o] - m[o] * sc);
    a0 += w2[o] * u;
    a1 += w2[64 + o] * u;
  }
  out[((size_t)b * 2 + 0) * HW + hw] = tanhf(a0);
  out[((size_t)b * 2 + 1) * HW + hw] = tanhf(a1);
}

// ---------------------------------------------------------------------------
extern "C" void kernel_launch(void* const* d_in, const int* in_sizes, int n_in,
                              void* d_out, int out_size, void* d_ws, size_t ws_size,
                              hipStream_t stream) {
  (void)in_sizes; (void)n_in; (void)out_size; (void)ws_size;
  const float* x       = (const float*)d_in[0];
  const float* convc_w = (const float*)d_in[1];
  const float* convc_b = (const float*)d_in[2];
  const float* convp_w = (const float*)d_in[3];
  const float* convp_b = (const float*)d_in[4];
  const float* convt_w = (const float*)d_in[5];
  const float* convt_b = (const float*)d_in[6];
  const float* tm_w    = (const float*)d_in[7];
  const float* tm_b    = (const float*)d_in[8];
  const float* tf_w    = (const float*)d_in[9];
  const float* tf_b    = (const float*)d_in[10];
  const float* poi_w   = (const float*)d_in[11];
  const float* poi_b   = (const float*)d_in[12];
  const float* bn1_g   = (const float*)d_in[13];
  const float* bn1_b   = (const float*)d_in[14];
  const float* bn1_m   = (const float*)d_in[15];
  const float* bn1_v   = (const float*)d_in[16];
  const float* conv1_w = (const float*)d_in[17];
  const float* conv1_b = (const float*)d_in[18];
  const float* rb1_g   = (const float*)d_in[19];
  const float* rb1_b   = (const float*)d_in[20];
  const float* rb1_m   = (const float*)d_in[21];
  const float* rb1_v   = (const float*)d_in[22];
  const float* rc1_w   = (const float*)d_in[23];
  const float* rc1_b   = (const float*)d_in[24];
  const float* rb2_g   = (const float*)d_in[25];
  const float* rb2_b   = (const float*)d_in[26];
  const float* rb2_m   = (const float*)d_in[27];
  const float* rb2_v   = (const float*)d_in[28];
  const float* rc2_w   = (const float*)d_in[29];
  const float* rc2_b   = (const float*)d_in[30];
  const float* plus_w  = (const float*)d_in[31];
  const float* plus_b  = (const float*)d_in[32];
  const float* bn2_g   = (const float*)d_in[33];
  const float* bn2_b   = (const float*)d_in[34];
  const float* bn2_m   = (const float*)d_in[35];
  const float* bn2_v   = (const float*)d_in[36];
  const float* conv2_w = (const float*)d_in[37];
  const float* conv2_b = (const float*)d_in[38];

  // Workspace layout (floats).  Total ~27.3 MB.
  float* W = (float*)d_ws;
  float* cpt    = W;                       // 1,048,576
  float* z2tp   = cpt    + 1048576;        // 1,048,576  (K/2 x 32 x 2 act)
  float* z1     = z2tp   + 1048576;        // 1,048,576  (64-ch stride, 56 used)
  float* z2part = z1     + 1048576;        // 1,048,576  (8 x 4096 x 32)
  float* z2out  = z2part + 1048576;        //   131,072  (4096 x 32)
  float* apad   = z2out  + 131072;         // 1,253,376  zero-padded relu(bn(cpt))
  float* z3pad  = apad   + 1253376;        // 1,253,376  zero-padded z3

  // Zero both padded buffers once; borders stay zero, interiors rewritten.
  {
    int n = 2 * 1253376;
    k_fill_zero<<<(n + 255) / 256, 256, 0, stream>>>(apad, n);
  }

  k_head<<<64, 256, 0, stream>>>(x, convc_w, convc_b, convp_w, convp_b,
      convt_w, convt_b, tm_w, tm_b, tf_w, tf_b, poi_w, poi_b,
      bn1_g, bn1_b, bn1_m, bn1_v, conv1_w, conv1_b, cpt);

  for (int it = 0; it < 4; ++it) {
    k_prep<<<4096, 256, 0, stream>>>(cpt, rb1_g, rb1_b, rb1_m, rb1_v, apad, z2tp);
    {
      dim3 g(NPLUS / 16, KSPLIT);
      k_plus_gemm<<<g, 32, 0, stream>>>(plus_w, z2tp, z2part);
    }
    k_plus_reduce<<<512, 256, 0, stream>>>(z2part, plus_b, z2out);
    {
      dim3 g(NB * HH, 4);
      k_conv3x3<<<g, 32, 0, stream>>>(apad, rc1_w, rc1_b, z1, nullptr, 56);
    }
    k_z3<<<4096, 256, 0, stream>>>(z1, z2out, rb2_g, rb2_b, rb2_m, rb2_v, z3pad);
    {
      dim3 g(NB * HH, 4);
      k_conv3x3<<<g, 32, 0, stream>>>(z3pad, rc2_w, rc2_b, cpt, cpt, 64);
    }
  }

  k_tail<<<64, 256, 0, stream>>>(cpt, bn2_g, bn2_b, bn2_m, bn2_v,
                                 conv2_w, conv2_b, (float*)d_out);
}